// FourierKernel_57664230916105
// MI455X (gfx1250) — compile-verified
//
#include <hip/hip_runtime.h>
#include <hip/hip_bf16.h>

// CDNA5 / gfx1250. wave32.
typedef __attribute__((ext_vector_type(2))) float v2f;
typedef __attribute__((ext_vector_type(8))) float v8f;

// ---------------------------------------------------------------------------
// Twiddle table: Wt[n][r], n in [0,512), r in [0,256).
//   r <  128 : cos(2*pi*r*n/512)        * filt[r]
//   r >= 128 : sin(2*pi*(r-128)*n/512)  * filt[r-128]
// Phase reduced exactly with integer mod 512 before float conversion.
// ---------------------------------------------------------------------------
__global__ void fk_twiddle(const float* __restrict__ filt, float* __restrict__ Wt) {
    int t = blockIdx.x * 256 + threadIdx.x;   // 512*256 = 131072 entries
    if (t >= 512 * 256) return;
    int n = t >> 8;
    int r = t & 255;
    int j = r & 127;
    int phase = (j * n) & 511;                 // exact argument reduction
    float theta = (float)phase * (6.2831853071795864769f / 512.0f);
    float s, c;
    sincosf(theta, &s, &c);
    Wt[t] = ((r < 128) ? c : s) * filt[j];
}

// ---------------------------------------------------------------------------
// Generic fp32 WMMA GEMM: C = A(MxK, row-major, lda) @ B(KxN, row-major, ldb)
//   block tile 128x128, K-chunk 32, 256 threads (8 wave32s).
//   Wave w owns the 16-wide N strip [w*16, w*16+16) and 8 M tiles.
//   mode 0: C[b] row-major (ldc), scaled.
//   mode 1: stage-1 "store Y transposed": global row i -> (b = i>>12, m = i&4095),
//           element written at C[b*(256*4096) + r*4096 + m].
// M, N must be multiples of 128; K multiple of 32 (true for all three calls).
// ---------------------------------------------------------------------------
__global__ __launch_bounds__(256, 2)
void fk_gemm_wmma(const float* __restrict__ A, const float* __restrict__ B,
                  float* __restrict__ C,
                  int lda, int ldb, int ldc, int K,
                  long long sA, long long sB, long long sC,
                  int mode, float scale) {
    __shared__ float As[128][36];   // 128 x (32+4)  : pad -> conflict-free frag reads
    __shared__ float Bs[32][144];   // 32  x (128+16): half-waves land 32 banks apart

    const int tid  = threadIdx.x;
    const int wave = tid >> 5;
    const int lane = tid & 31;
    const int bz   = blockIdx.z;

    const float* Ab = A + (long long)bz * sA + (size_t)blockIdx.y * 128 * lda;
    const float* Bb = B + (long long)bz * sB + (size_t)blockIdx.x * 128;

    v8f acc[8] = {};                // 8 M-tiles x 16x16 fp32 accum (64 VGPRs)

    const int am = lane & 15;             // M (and N) index within tile
    const int hk = (lane >> 4) << 1;      // K half-wave offset (0 or 2)

    for (int kb = 0; kb < K; kb += 32) {
        // ---- stage 128x32 of A and 32x128 of B into LDS (float4, coalesced) ----
        #pragma unroll
        for (int i = 0; i < 4; ++i) {
            int idx = i * 256 + tid;
            int m  = idx >> 3, kq = (idx & 7) << 2;          // A: 8 float4 per row
            float4 av = *(const float4*)(Ab + (size_t)m * lda + kb + kq);
            *(float4*)&As[m][kq] = av;
            int kk = idx >> 5, nq = (idx & 31) << 2;         // B: 32 float4 per row
            float4 bv = *(const float4*)(Bb + (size_t)(kb + kk) * ldb + nq);
            *(float4*)&Bs[kk][nq] = bv;
        }
        // prefetch next K-chunk while we compute on this one
        if (kb + 32 < K) {
            __builtin_prefetch(Ab + (size_t)(tid >> 1) * lda + kb + 32, 0, 1);
            __builtin_prefetch(Bb + (size_t)(kb + 32 + (tid >> 3)) * ldb + ((tid & 7) << 4), 0, 1);
        }
        __syncthreads();

        // ---- 8 k-steps x 8 M-tiles of V_WMMA_F32_16X16X4_F32 ----
        #pragma unroll
        for (int ks = 0; ks < 32; ks += 4) {
            const int k0 = ks + hk;       // ISA 7.12.2: k = 2*(lane>>4) + vgpr
            v2f bf;
            bf.x = Bs[k0][wave * 16 + am];
            bf.y = Bs[k0 + 1][wave * 16 + am];
            #pragma unroll
            for (int t = 0; t < 8; ++t) {
                v2f af = *(const v2f*)&As[t * 16 + am][k0];   // contiguous 8B
                acc[t] = __builtin_amdgcn_wmma_f32_16x16x4_f32(
                    false, af, false, bf, (short)0, acc[t], false, false);
            }
        }
        __syncthreads();
    }

    // ---- store: D layout (7.12.2): m = v + 8*(lane>>4), n = lane&15 ----
    const int m_h = (lane >> 4) << 3;
    if (mode == 0) {
        float* Cb = C + (long long)bz * sC;
        #pragma unroll
        for (int t = 0; t < 8; ++t) {
            const int gm0 = blockIdx.y * 128 + t * 16 + m_h;
            const int gn  = blockIdx.x * 128 + wave * 16 + am;
            #pragma unroll
            for (int v = 0; v < 8; ++v)
                Cb[(size_t)(gm0 + v) * ldc + gn] = acc[t][v] * scale;
        }
    } else {
        #pragma unroll
        for (int t = 0; t < 8; ++t) {
            const int gi0 = blockIdx.y * 128 + t * 16 + m_h;   // global Y row
            const int r   = blockIdx.x * 128 + wave * 16 + am; // freq index
            #pragma unroll
            for (int v = 0; v < 8; ++v) {
                const int gi = gi0 + v;
                size_t idx = ((size_t)(gi >> 12) << 20)        // batch*(256*4096)
                           + (size_t)r * 4096 + (gi & 4095);
                C[idx] = acc[t][v] * scale;
            }
        }
    }
}

// ---------------------------------------------------------------------------
// d_in[0]=x (4,4096,512) f32, d_in[1]=y (4,4096,512) f32, d_in[2]=filt (128) f32
// d_out: (4,4096,4096) f32.  Workspace: Wt 512KB + Zx 16MB + Zyt 16MB ~ 34MB.
// ---------------------------------------------------------------------------
extern "C" void kernel_launch(void* const* d_in, const int* in_sizes, int n_in,
                              void* d_out, int out_size, void* d_ws, size_t ws_size,
                              hipStream_t stream) {
    const float* x    = (const float*)d_in[0];
    const float* y    = (const float*)d_in[1];
    const float* filt = (const float*)d_in[2];
    float* out = (float*)d_out;

    float* Wt  = (float*)d_ws;                 // 512*256
    float* Zx  = Wt + 512 * 256;               // 16384*256
    float* Zyt = Zx + 16384 * 256;             // 4*256*4096

    fk_twiddle<<<512, 256, 0, stream>>>(filt, Wt);

    dim3 blk(256);
    // Zx = x @ Wt, scaled by 1/rank (folds the final /128 into the X side)
    fk_gemm_wmma<<<dim3(2, 128, 1), blk, 0, stream>>>(
        x, Wt, Zx, 512, 256, 256, 512, 0, 0, 0, /*mode=*/0, 1.0f / 128.0f);
    // Zyt = (y @ Wt)^T per batch  -> row-major (b, 256, 4096)
    fk_gemm_wmma<<<dim3(2, 128, 1), blk, 0, stream>>>(
        y, Wt, Zyt, 512, 256, 0, 512, 0, 0, 0, /*mode=*/1, 1.0f);
    // out[b] = Zx[b] @ Zyt[b]
    fk_gemm_wmma<<<dim3(32, 32, 4), blk, 0, stream>>>(
        Zx, Zyt, out, 256, 4096, 4096, 256,
        (long long)4096 * 256, (long long)256 * 4096, (long long)4096 * 4096,
        /*mode=*/0, 1.0f);
}